// FPSKNNGrouper_41575283425634
// MI455X (gfx1250) — compile-verified
//
#include <hip/hip_runtime.h>
#include <hip/hip_bf16.h>

#define BATCH   16
#define NPTS    4096
#define NCH     32
#define NPOINT  1024
#define KNN     16
#define NB      256          // columns per chunk in knn kernel

typedef __attribute__((ext_vector_type(2))) float v2f;
typedef __attribute__((ext_vector_type(8))) float v8f;

// ---------------------------------------------------------------------------
// Kernel 1: farthest point sampling. One workgroup per batch, 512 threads,
// 8 points per thread held in registers; 1024 serial argmax steps.
// key = (dist_bits << 32) | (~idx)  -> max key == max dist, tie -> lowest idx
// Second reduction stage runs on wave 0 only to shorten the serial chain.
// ---------------------------------------------------------------------------
__global__ __launch_bounds__(512) void fps_kernel(const float* __restrict__ x,
                                                  int* __restrict__ cent) {
    __shared__ float sxyz[NPTS * 3];               // 48 KB
    __shared__ unsigned long long sred[16];        // per-wave partial maxes
    __shared__ unsigned long long sfin;            // broadcast slot

    const int b = blockIdx.x;
    const int t = threadIdx.x;
    const int lane = t & 31;
    const int wv = t >> 5;
    const float* xb = x + (size_t)b * NPTS * NCH;

    for (int i = t; i < NPTS; i += 512) {
        sxyz[i * 3 + 0] = xb[i * NCH + 0];
        sxyz[i * 3 + 1] = xb[i * NCH + 1];
        sxyz[i * 3 + 2] = xb[i * NCH + 2];
    }
    __syncthreads();

    float px[8], py[8], pz[8], pd[8];
#pragma unroll
    for (int j = 0; j < 8; ++j) {
        const int i = t * 8 + j;
        px[j] = sxyz[i * 3 + 0];
        py[j] = sxyz[i * 3 + 1];
        pz[j] = sxyz[i * 3 + 2];
        pd[j] = 1e10f;
    }

    int far = 0;
    for (int it = 0; it < NPOINT; ++it) {
        if (t == 0) cent[b * NPOINT + it] = far;
        const float cx = sxyz[far * 3 + 0];
        const float cy = sxyz[far * 3 + 1];
        const float cz = sxyz[far * 3 + 2];

        unsigned long long best = 0ull;
#pragma unroll
        for (int j = 0; j < 8; ++j) {
            const float dx = px[j] - cx, dy = py[j] - cy, dz = pz[j] - cz;
            const float d = dx * dx + dy * dy + dz * dz;
            pd[j] = fminf(pd[j], d);
            const unsigned inv = 0xFFFFFFFFu - (unsigned)(t * 8 + j);
            const unsigned long long key =
                ((unsigned long long)__float_as_uint(pd[j]) << 32) | inv;
            best = key > best ? key : best;
        }
        // intra-wave max (wave32)
#pragma unroll
        for (int off = 16; off > 0; off >>= 1) {
            const unsigned long long o = __shfl_xor(best, off, 32);
            best = o > best ? o : best;
        }
        if (lane == 0) sred[wv] = best;
        __syncthreads();
        if (wv == 0) {
            unsigned long long g = sred[lane & 15];
#pragma unroll
            for (int off = 8; off > 0; off >>= 1) {
                const unsigned long long o = __shfl_xor(g, off, 32);
                g = o > g ? o : g;
            }
            if (lane == 0) sfin = g;
        }
        __syncthreads();
        far = (int)(0xFFFFFFFFu - (unsigned)(sfin & 0xFFFFFFFFull));
    }
}

// ---------------------------------------------------------------------------
// Kernel 2: distances via V_WMMA_F32_16X16X4_F32, per-row top-16, gather.
// Grid: BATCH * (NPOINT/16) workgroups, 512 threads = 16 waves.
// Wave w: computes column-block w of each 16x256 distance chunk (1 WMMA),
// then owns row w for top-k selection and the final output gather.
// ---------------------------------------------------------------------------
__global__ __launch_bounds__(512) void knn_kernel(const float* __restrict__ x,
                                                  const int* __restrict__ cent,
                                                  float* __restrict__ out) {
    // 64 KB: merge buffer (16 waves * 32 lanes * 16 u64); front 16 KB doubles
    // as the 16x256 f32 distance tile during the main loop.
    __shared__ unsigned long long smrg[16 * 32 * 16];
    float* sdist = (float*)smrg;

    const int b    = blockIdx.x >> 6;       // 64 row-tiles per batch
    const int tile = blockIdx.x & 63;
    const int t    = threadIdx.x;
    const int w    = t >> 5;                // wave id 0..15
    const int lane = t & 31;
    const int lh   = lane & 15;
    const int hi   = lane >> 4;             // 0 = lanes 0-15, 1 = lanes 16-31
    const float* xb = x + (size_t)b * NPTS * NCH;

    // ---- A fragment: row m = tile*16 + lh, K = {-2x,-2y,-2z,s2} ----
    const int m = tile * 16 + lh;
    const int sidx = cent[b * NPOINT + m];
    const float sx = xb[sidx * NCH + 0];
    const float sy = xb[sidx * NCH + 1];
    const float sz = xb[sidx * NCH + 2];
    const float s2 = sx * sx + sy * sy + sz * sz;
    v2f afrag;
    afrag.x = hi ? (-2.0f * sz) : (-2.0f * sx);   // V0 = {K0 | K2}
    afrag.y = hi ? s2           : (-2.0f * sy);   // V1 = {K1 | K3}

    // per-lane sorted (ascending) top-16 of keys (dist_bits<<32 | col_idx)
    unsigned long long topa[KNN];
#pragma unroll
    for (int k = 0; k < KNN; ++k) topa[k] = ~0ull;

    for (int ch = 0; ch < NPTS / NB; ++ch) {          // 16 chunks
        // ---- B fragment: column n, K = {x,y,z,1} ----
        const int n = ch * NB + w * 16 + lh;
        const float cx = xb[n * NCH + 0];
        const float cy = xb[n * NCH + 1];
        const float cz = xb[n * NCH + 2];
        const float d2 = cx * cx + cy * cy + cz * cz;
        v2f bfrag;
        bfrag.x = hi ? cz   : cx;
        bfrag.y = hi ? 1.0f : cy;

        v8f acc = {};
        acc = __builtin_amdgcn_wmma_f32_16x16x4_f32(
            /*neg_a=*/false, afrag, /*neg_b=*/false, bfrag,
            /*c_mod=*/(short)0, acc, /*reuse_a=*/false, /*reuse_b=*/false);

        // C/D layout: VGPR j -> row j + 8*hi, col = lane%16
#pragma unroll
        for (int j = 0; j < 8; ++j) {
            const int row = j + 8 * hi;
            sdist[row * NB + w * 16 + lh] = acc[j] + d2;
        }
        __syncthreads();

        // ---- top-k scan: wave w owns row w; lane covers cols lane+32j ----
#pragma unroll
        for (int j = 0; j < NB / 32; ++j) {
            const int col = lane + 32 * j;
            float d = sdist[w * NB + col];
            d = fmaxf(d, 0.0f);                       // guard fp cancellation
            const unsigned idx = (unsigned)(ch * NB + col);
            const unsigned long long key =
                ((unsigned long long)__float_as_uint(d) << 32) | idx;
            // whole-wave skip: if no lane can improve its top-16, bypass the
            // ~80-instruction insert block (dominant VALU cost).
            if (__any(key < topa[KNN - 1])) {
                // branch-free sorted insert (static indices after unroll)
#pragma unroll
                for (int k = KNN - 1; k >= 1; --k) {
                    const bool lt  = key < topa[k];
                    const bool ltp = key < topa[k - 1];
                    topa[k] = lt ? (ltp ? topa[k - 1] : key) : topa[k];
                }
                topa[0] = (key < topa[0]) ? key : topa[0];
            }
        }
        __syncthreads();   // before next chunk overwrites sdist
    }

    // ---- merge 32 sorted per-lane lists -> global top-16, gather output ----
    unsigned long long* wbuf = smrg + w * (32 * KNN);
#pragma unroll
    for (int k = 0; k < KNN; ++k) wbuf[lane * KNN + k] = topa[k];
    __syncthreads();

    int head = 0;
    float* outrow = out + (((size_t)b * NPOINT + (size_t)(tile * 16 + w)) * KNN) * NCH;
    for (int k = 0; k < KNN; ++k) {
        const unsigned long long mykey = wbuf[lane * KNN + head];
        unsigned long long mn = mykey;
#pragma unroll
        for (int off = 16; off > 0; off >>= 1) {
            const unsigned long long o = __shfl_xor(mn, off, 32);
            mn = o < mn ? o : mn;
        }
        if (mykey == mn) ++head;                       // pop winner
        const unsigned nidx = (unsigned)(mn & 0xFFFFFFFFull);
        // 32 channels, one per lane: coalesced 128B gather + store
        outrow[k * NCH + lane] = xb[(size_t)nidx * NCH + lane];
    }
}

// ---------------------------------------------------------------------------
extern "C" void kernel_launch(void* const* d_in, const int* in_sizes, int n_in,
                              void* d_out, int out_size, void* d_ws, size_t ws_size,
                              hipStream_t stream) {
    const float* x = (const float*)d_in[0];          // [16, 4096, 32] f32
    float* out = (float*)d_out;                      // [16, 1024, 16, 32] f32
    int* cent = (int*)d_ws;                          // [16, 1024] int32

    fps_kernel<<<BATCH, 512, 0, stream>>>(x, cent);
    knn_kernel<<<BATCH * (NPOINT / 16), 512, 0, stream>>>(x, cent, out);
}